// KanOutPutModule_91010357002569
// MI455X (gfx1250) — compile-verified
//
#include <hip/hip_runtime.h>

typedef __attribute__((ext_vector_type(2))) float v2f;
typedef __attribute__((ext_vector_type(8))) float v8f;

// Grid: pts[i] = 0.4*i - 2.2, i = 0..11  (G=5, K=3, h=0.4). 8 cubic bases.
__device__ __forceinline__ float kan_epilogue(float dot, int a, float bm,
                                              const float* __restrict__ coef,
                                              const float* __restrict__ scale_base,
                                              const float* __restrict__ scale_sp)
{
    // s = silu(h)
    float h = dot + bm;
    float s = h * (1.0f / (1.0f + __expf(-h)));
    // silu(s) again per reference: scale_base * silu(xa)
    float silu_s = s * (1.0f / (1.0f + __expf(-s)));

    // Uniform cubic B-spline, closed form. t in [0,11) <=> s in [-2.2, 2.2)
    float t  = (s + 2.2f) * 2.5f;
    float fi = floorf(t);
    int   i  = (int)fi;
    float spline = 0.0f;
    if (i >= 0 && i <= 10) {
        float u   = t - fi;
        float omu = 1.0f - u;
        float u2  = u * u;
        float u3  = u2 * u;
        float w0 = omu * omu * omu * (1.0f / 6.0f);
        float w1 = (3.0f * u3 - 6.0f * u2 + 4.0f) * (1.0f / 6.0f);
        float w3 = u3 * (1.0f / 6.0f);
        float w2 = 1.0f - w0 - w1 - w3;
        float wts[4] = {w0, w1, w2, w3};
        const float* ca = coef + a * 8;
#pragma unroll
        for (int d = 0; d < 4; ++d) {
            int c = i - 3 + d;                  // basis index, clip to [0,8)
            if (c >= 0 && c < 8) spline += wts[d] * ca[c];
        }
    }
    return scale_base[a] * silu_s + scale_sp[a] * spline;
}

// Dynamic (per-lane) select of one element from the WMMA accumulator.
__device__ __forceinline__ float acc_select(const v8f& acc, int m8)
{
    float r = acc[0];
#pragma unroll
    for (int v = 1; v < 8; ++v) r = (m8 == v) ? acc[v] : r;
    return r;
}

__global__ __launch_bounds__(256) void kan_out_kernel(
    const float* __restrict__ x,          // (B, 64, 128)
    const float* __restrict__ w_mlp,      // (128)
    const float* __restrict__ b_mlp,      // (1)
    const float* __restrict__ coef,       // (64, 8)
    const float* __restrict__ scale_base, // (64)
    const float* __restrict__ scale_sp,   // (64)
    const float* __restrict__ kan_bias,   // (1)
    float* __restrict__ out,              // (B)
    int nB)
{
    const int lane = threadIdx.x & 31;
    const int gw   = blockIdx.x * 8 + (threadIdx.x >> 5);  // one wave per b
    if (gw >= nB) return;                                  // wave-uniform branch

    const int hi = lane >> 4;      // 0: K-slots 0,1 (k in [0,64)); 1: slots 2,3 (k in [64,128))
    const int m  = lane & 15;      // row within 16-row tile

    // Per-lane base: row = gw*64 + tile*16 + m, column base = hi*64 floats.
    const float* xbase = x + ((size_t)gw * 64 + (size_t)m) * 128 + (size_t)hi * 64;
    const float4* xt0 = (const float4*)(xbase);
    const float4* xt1 = (const float4*)(xbase + 16 * 128);
    const float4* xt2 = (const float4*)(xbase + 32 * 128);
    const float4* xt3 = (const float4*)(xbase + 48 * 128);
    const float4* wv  = (const float4*)w_mlp + hi * 16;    // lane-group's half of w

    v8f acc0 = {0.f, 0.f, 0.f, 0.f, 0.f, 0.f, 0.f, 0.f};
    v8f acc1 = acc0, acc2 = acc0, acc3 = acc0;

#pragma unroll
    for (int j = 0; j < 16; ++j) {
        float4 wq = wv[j];
        v2f b0 = {wq.x, wq.y};
        v2f b1 = {wq.z, wq.w};
        float4 q0 = xt0[j];
        float4 q1 = xt1[j];
        float4 q2 = xt2[j];
        float4 q3 = xt3[j];
        v2f a;
        a = (v2f){q0.x, q0.y};
        acc0 = __builtin_amdgcn_wmma_f32_16x16x4_f32(false, a, false, b0, (short)0, acc0, false, false);
        a = (v2f){q0.z, q0.w};
        acc0 = __builtin_amdgcn_wmma_f32_16x16x4_f32(false, a, false, b1, (short)0, acc0, false, false);
        a = (v2f){q1.x, q1.y};
        acc1 = __builtin_amdgcn_wmma_f32_16x16x4_f32(false, a, false, b0, (short)0, acc1, false, false);
        a = (v2f){q1.z, q1.w};
        acc1 = __builtin_amdgcn_wmma_f32_16x16x4_f32(false, a, false, b1, (short)0, acc1, false, false);
        a = (v2f){q2.x, q2.y};
        acc2 = __builtin_amdgcn_wmma_f32_16x16x4_f32(false, a, false, b0, (short)0, acc2, false, false);
        a = (v2f){q2.z, q2.w};
        acc2 = __builtin_amdgcn_wmma_f32_16x16x4_f32(false, a, false, b1, (short)0, acc2, false, false);
        a = (v2f){q3.x, q3.y};
        acc3 = __builtin_amdgcn_wmma_f32_16x16x4_f32(false, a, false, b0, (short)0, acc3, false, false);
        a = (v2f){q3.z, q3.w};
        acc3 = __builtin_amdgcn_wmma_f32_16x16x4_f32(false, a, false, b1, (short)0, acc3, false, false);
    }

    // D[m][n] is identical across n (all B columns carry w), so:
    //   lanes 0-15:  acc[v] = dot(row v)      of the tile
    //   lanes 16-31: acc[v] = dot(row v + 8)
    // Lanes 0-7 handle rows 0-7, lanes 16-23 handle rows 8-15 (others idle).
    const float bm = b_mlp[0];
    const int  m8     = lane & 7;
    const bool active = (lane & 8) == 0;
    float total = 0.0f;
    if (active) {
        const int mrow = hi * 8 + m8;        // row within tile = a % 16
        total += kan_epilogue(acc_select(acc0, m8), mrow +  0, bm, coef, scale_base, scale_sp);
        total += kan_epilogue(acc_select(acc1, m8), mrow + 16, bm, coef, scale_base, scale_sp);
        total += kan_epilogue(acc_select(acc2, m8), mrow + 32, bm, coef, scale_base, scale_sp);
        total += kan_epilogue(acc_select(acc3, m8), mrow + 48, bm, coef, scale_base, scale_sp);
    }

    // Wave32 reduction over the 16 active lanes (inactive contribute 0).
#pragma unroll
    for (int off = 16; off > 0; off >>= 1)
        total += __shfl_down(total, off, 32);

    if (lane == 0) out[gw] = total + kan_bias[0];
}

extern "C" void kernel_launch(void* const* d_in, const int* in_sizes, int n_in,
                              void* d_out, int out_size, void* d_ws, size_t ws_size,
                              hipStream_t stream)
{
    const float* x          = (const float*)d_in[0];
    const float* w_mlp      = (const float*)d_in[1];
    const float* b_mlp      = (const float*)d_in[2];
    const float* coef       = (const float*)d_in[3];
    const float* scale_base = (const float*)d_in[4];
    const float* scale_sp   = (const float*)d_in[5];
    const float* kan_bias   = (const float*)d_in[6];
    float* out = (float*)d_out;

    const int nB = out_size;                 // 32768 output rows (B, 1)
    const int blocks = (nB + 7) / 8;         // 8 waves per 256-thread block, 1 wave per b
    kan_out_kernel<<<blocks, 256, 0, stream>>>(x, w_mlp, b_mlp, coef,
                                               scale_base, scale_sp, kan_bias,
                                               out, nB);
}